// XGBUncertainity_85349590106795
// MI455X (gfx1250) — compile-verified
//
#include <hip/hip_runtime.h>
#include <hip/hip_bf16.h>

typedef __attribute__((ext_vector_type(2))) float v2f;
typedef __attribute__((ext_vector_type(8))) float v8f;

#define M_TOTAL   16384   // nr*nb
#define N_TRAIN   8192
#define HALF_N    4096    // train points per wave-group (n split in 2 per WG)
#define DIMS      32
#define TN        128     // train tile (points) per buffer per iteration
#define TILES     (HALF_N / TN)
#define TSTRIDE   36      // padded LDS stride in floats (float4-aligned, conflict-free b64)
#define EPSF      1e-8f

// ---------------------------------------------------------------------------
// Kernel 1: Scott's-rule bandwidth from train_x -> ws[0] = 1/(2*bw^2)
// ---------------------------------------------------------------------------
__global__ __launch_bounds__(1024) void bw_kernel(const float* __restrict__ tx,
                                                  float* __restrict__ ws) {
  __shared__ float s1[32][33];
  __shared__ float s2[32][33];
  const int t   = threadIdx.x;
  const int col = t & 31;
  const int seg = t >> 5;          // 32 segments of 256 rows
  float a1 = 0.0f, a2 = 0.0f;
  const int r0 = seg * 256;
  for (int r = r0; r < r0 + 256; ++r) {
    float v = tx[r * DIMS + col];
    a1 += v;
    a2 += v * v;
  }
  s1[seg][col] = a1;
  s2[seg][col] = a2;
  __syncthreads();
  if (t < 32) {                    // one wave32 finishes the reduction
    float b1 = 0.0f, b2 = 0.0f;
    for (int g = 0; g < 32; ++g) { b1 += s1[g][t]; b2 += s2[g][t]; }
    const float n = (float)N_TRAIN;
    float var = (b2 - b1 * b1 / n) / (n - 1.0f);   // ddof = 1
    float sd  = sqrtf(fmaxf(var, 0.0f));
    for (int off = 1; off < 32; off <<= 1) sd += __shfl_xor(sd, off, 32);
    if (t == 0) {
      float bw = (sd / 32.0f) * powf((float)N_TRAIN, -1.0f / (float)(DIMS + 4));
      ws[0] = 1.0f / (2.0f * bw * bw + 1e-30f);
    }
  }
}

// ---------------------------------------------------------------------------
// Kernel 2: fused cdist^2 -> exp -> Nadaraya-Watson ratio, GEMM on WMMA f32.
// Grid: 256 WGs x 256 threads (8 waves). WG owns 64 query rows; waves 0-3 and
// 4-7 process disjoint halves of the train set (double-buffered LDS tiles)
// and partial sums are combined through LDS. 2048 waves total for occupancy.
// ---------------------------------------------------------------------------
__global__ __launch_bounds__(256) void kde_wmma_kernel(
    const float* __restrict__ xq,   // [M_TOTAL, 32]
    const float* __restrict__ tx,   // [N_TRAIN, 32]
    const float* __restrict__ tr,   // [N_TRAIN]
    const float* __restrict__ ws,   // ws[0] = inv_bw_sq
    float* __restrict__ rx) {       // [M_TOTAL]
  __shared__ float tile[2][TN * TSTRIDE];  // 2 x 18432 B train tiles (padded)
  __shared__ float tnorm[2][TN];           // |t|^2 per tile point
  __shared__ float rres[2][TN];            // residuals per tile point
  __shared__ float qn_lds[64];             // |q|^2 for the WG's 64 rows
  __shared__ float cnum[8][16];            // per-wave partial numerators
  __shared__ float cden[8][16];            // per-wave partial denominators

  const int tid   = threadIdx.x;
  const int wave  = tid >> 5;
  const int lane  = tid & 31;
  const int laneN = lane & 15;
  const int half  = lane >> 4;          // 0: lanes 0-15, 1: lanes 16-31
  const int rgrp  = wave & 3;           // which 16-row tile of the WG's 64 rows
  const int nsel  = wave >> 2;          // which train half (0 or 1)
  const int m0    = blockIdx.x * 64 + rgrp * 16;
  const float inv_bw = ws[0];

  // --- A operand: 16x32 fp32 queries, 8 K=4 steps, resident in VGPRs ------
  // ISA 32-bit A 16x4 layout: lane holds row M=lane%16; lanes 0-15 carry
  // k-pair {4s,4s+1}, lanes 16-31 carry {4s+2,4s+3}.
  v2f a[8];
  const float* qptr = xq + (m0 + laneN) * DIMS + half * 2;
  float qh = 0.0f;
#pragma unroll
  for (int s = 0; s < 8; ++s) {
    a[s] = *(const v2f*)(qptr + 4 * s);
    qh += a[s].x * a[s].x + a[s].y * a[s].y;
  }
  // each half-wave holds half the norm; xor-16 completes it
  float qfull = qh + __shfl_xor(qh, 16, 32);
  if (nsel == 0 && half == 0) qn_lds[rgrp * 16 + laneN] = qfull;
  __syncthreads();

  // C-layout: element e of lane L is row M = e + 8*half, col N = L%16
  float qe[8];
#pragma unroll
  for (int e = 0; e < 8; ++e) qe[e] = qn_lds[rgrp * 16 + e + half * 8];

  float num[8], den[8];
#pragma unroll
  for (int e = 0; e < 8; ++e) { num[e] = 0.0f; den[e] = 0.0f; }

#pragma unroll 1
  for (int it = 0; it < TILES; ++it) {
    __syncthreads();                      // previous tiles fully consumed
    {
      // all 256 threads load: one thread per train point, both buffers
      const int buf = tid >> 7;           // 0: first half, 1: second half
      const int p   = tid & 127;
      const int gp  = buf * HALF_N + it * TN + p;
      const float4* src = (const float4*)(tx + gp * DIMS);
      float s = 0.0f;
#pragma unroll
      for (int i = 0; i < 8; ++i) {
        float4 v = src[i];
        *(float4*)&tile[buf][p * TSTRIDE + 4 * i] = v;
        s += v.x * v.x + v.y * v.y + v.z * v.z + v.w * v.w;
      }
      tnorm[buf][p] = s;
      rres[buf][p]  = tr[gp];
      if (it + 1 < TILES)                 // prefetch own next-tile row
        __builtin_prefetch(tx + (gp + TN) * DIMS, 0, 0);
    }
    __syncthreads();

#pragma unroll 1
    for (int nt = 0; nt < TN / 16; ++nt) {
      const int N = nt * 16 + laneN;
      // B operand mirrors A: lane holds col N; halves carry k-pairs
      const float* bp = &tile[nsel][N * TSTRIDE + half * 2];
      v8f c = {};
#pragma unroll
      for (int s = 0; s < 8; ++s) {
        v2f b = *(const v2f*)(bp + 4 * s);
        c = __builtin_amdgcn_wmma_f32_16x16x4_f32(
            /*neg_a=*/false, a[s], /*neg_b=*/false, b,
            /*c_mod=*/(short)0, c, /*reuse_a=*/false, /*reuse_b=*/false);
      }
      const float tn = tnorm[nsel][N];
      const float rn = rres[nsel][N];
#pragma unroll
      for (int e = 0; e < 8; ++e) {
        float d2 = qe[e] + tn - 2.0f * c[e];
        d2 = fmaxf(d2, 0.0f);
        float w = __expf(-d2 * inv_bw);   // v_exp_f32 on VALU, overlaps WMMA
        num[e] += w * rn;
        den[e] += w;
      }
    }
  }

  // cross-lane reduce over the 16 columns held within each half-wave
#pragma unroll
  for (int e = 0; e < 8; ++e) {
    float nu = num[e], de = den[e];
#pragma unroll
    for (int off = 1; off < 16; off <<= 1) {
      nu += __shfl_xor(nu, off, 32);
      de += __shfl_xor(de, off, 32);
    }
    if (laneN == 0) {  // lane 0 -> rows m0+e ; lane 16 -> rows m0+8+e
      cnum[wave][e + half * 8] = nu;
      cden[wave][e + half * 8] = de;
    }
  }
  __syncthreads();

  // combine the two train-half partials; EPS added once on the full sum
  if (tid < 64) {
    const int wt = tid >> 4;              // row tile 0..3
    const int j  = tid & 15;              // row within tile
    float nu = cnum[wt][j] + cnum[wt + 4][j];
    float de = cden[wt][j] + cden[wt + 4][j];
    rx[blockIdx.x * 64 + wt * 16 + j] = nu / (de + EPSF);
  }
}

// ---------------------------------------------------------------------------
// Kernel 3: global min/max of rx and sigma -> ws[1..4]
// ---------------------------------------------------------------------------
__global__ __launch_bounds__(1024) void minmax_kernel(const float* __restrict__ rx,
                                                      const float* __restrict__ sig,
                                                      float* __restrict__ ws) {
  __shared__ float mn1[1024], mx1[1024], mn2[1024], mx2[1024];
  const int t = threadIdx.x;
  float a = 3.4e38f, b = -3.4e38f, c = 3.4e38f, d = -3.4e38f;
  for (int i = t; i < M_TOTAL; i += 1024) {
    float v = rx[i];  a = fminf(a, v); b = fmaxf(b, v);
    float u = sig[i]; c = fminf(c, u); d = fmaxf(d, u);
  }
  mn1[t] = a; mx1[t] = b; mn2[t] = c; mx2[t] = d;
  __syncthreads();
  for (int s = 512; s > 0; s >>= 1) {
    if (t < s) {
      mn1[t] = fminf(mn1[t], mn1[t + s]);
      mx1[t] = fmaxf(mx1[t], mx1[t + s]);
      mn2[t] = fminf(mn2[t], mn2[t + s]);
      mx2[t] = fmaxf(mx2[t], mx2[t + s]);
    }
    __syncthreads();
  }
  if (t == 0) { ws[1] = mn1[0]; ws[2] = mx1[0]; ws[3] = mn2[0]; ws[4] = mx2[0]; }
}

// ---------------------------------------------------------------------------
// Kernel 4: T = 0.5*norm(rx) + 0.5*norm(sigma)
// ---------------------------------------------------------------------------
__global__ __launch_bounds__(256) void combine_kernel(const float* __restrict__ rx,
                                                      const float* __restrict__ sig,
                                                      const float* __restrict__ ws,
                                                      float* __restrict__ out) {
  const int i = blockIdx.x * 256 + threadIdx.x;
  if (i < M_TOTAL) {
    const float rmin = ws[1], rmax = ws[2], smin = ws[3], smax = ws[4];
    float t1 = 0.5f * (rx[i]  - rmin) / (rmax - rmin + EPSF);
    float t2 = 0.5f * (sig[i] - smin) / (smax - smin + EPSF);
    out[i] = t1 + t2;
  }
}

// ---------------------------------------------------------------------------
extern "C" void kernel_launch(void* const* d_in, const int* in_sizes, int n_in,
                              void* d_out, int out_size, void* d_ws, size_t ws_size,
                              hipStream_t stream) {
  const float* x    = (const float*)d_in[0];   // [8,2048,32]
  const float* tx   = (const float*)d_in[1];   // [8192,32]
  const float* tr   = (const float*)d_in[2];   // [8192]
  const float* sig  = (const float*)d_in[3];   // [8,2048]
  float*       out  = (float*)d_out;           // [8,2048]
  float*       wsf  = (float*)d_ws;            // [0]=inv_bw_sq, [1..4]=minmax
  float*       rx   = wsf + 16;                // 16384 floats of scratch

  bw_kernel      <<<1,            1024, 0, stream>>>(tx, wsf);
  kde_wmma_kernel<<<M_TOTAL / 64,  256, 0, stream>>>(x, tx, tr, wsf, rx);
  minmax_kernel  <<<1,            1024, 0, stream>>>(rx, sig, wsf);
  combine_kernel <<<M_TOTAL / 256, 256, 0, stream>>>(rx, sig, wsf, out);
}